// OriCIFPhi_5428838662169
// MI455X (gfx1250) — compile-verified
//
#include <hip/hip_runtime.h>
#include <hip/hip_bf16.h>

#define BETA_F 1.0f
#define TAIL_THRES_F 0.5f
#define KC 64            // sources per LDS weight chunk
#define WSTRIDE (KC + 2) // padded LDS stride -> conflict-free ds_load_b64
#define NB 16            // batch (fixed by reference)

typedef __attribute__((ext_vector_type(2))) float v2f;
typedef __attribute__((ext_vector_type(8))) float v8f;

struct CifParams {
    int   Tpre;          // max pre-extension length over batches (atomicMax)
    int   pad;
    float total[NB];
    int   fl_pre[NB];
    float scale[NB];
    int   fl_final[NB];
};

static __device__ __forceinline__ int imin(int a, int b) { return a < b ? a : b; }
static __device__ __forceinline__ int imax(int a, int b) { return a > b ? a : b; }

__global__ void cif_init(CifParams* p) {
    if (threadIdx.x == 0) p->Tpre = 0;
}

// ---------------- Kernel 1: masked prefix-sum of alphas per batch ----------
__global__ __launch_bounds__(256) void cif_scan(const float* __restrict__ alphas,
                                                const unsigned char* __restrict__ mask,
                                                float* __restrict__ csum,
                                                CifParams* __restrict__ p, int S) {
    const int b = blockIdx.x, tid = threadIdx.x;
    const float* aB = alphas + (size_t)b * S;
    const unsigned char* mB = mask + (size_t)b * S;
    float* cB = csum + (size_t)b * S;

    const int PER = 16;                 // S = 4096 = 256*16
    int base = tid * PER;
    float loc[PER];
    float run = 0.0f;
#pragma unroll
    for (int j = 0; j < PER; ++j) {
        int s = base + j;
        float a = (s < S) ? aB[s] : 0.0f;
        if (s < S && mB[s]) a = 0.0f;
        run += a;
        loc[j] = run;
    }
    __shared__ float sd[256];
    sd[tid] = run;
    __syncthreads();
    for (int off = 1; off < 256; off <<= 1) {
        float v = (tid >= off) ? sd[tid - off] : 0.0f;
        __syncthreads();
        sd[tid] += v;
        __syncthreads();
    }
    float excl = sd[tid] - run;
#pragma unroll
    for (int j = 0; j < PER; ++j) {
        int s = base + j;
        if (s < S) cB[s] = excl + loc[j];
    }
    if (tid == 0) {
        float tot = sd[255];
        int fl = (int)floorf(tot / BETA_F);
        p->total[b]  = tot;
        p->fl_pre[b] = fl;
        atomicMax(&p->Tpre, fl);
    }
}

// ---------------- Kernel 2: tail weight reduction + scale/length -----------
__global__ __launch_bounds__(256) void cif_tail(const float* __restrict__ csum,
                                                CifParams* __restrict__ p,
                                                float* __restrict__ lens_out,
                                                int S) {
    const int b = blockIdx.x, tid = threadIdx.x;
    const float* cB = csum + (size_t)b * S;
    const int Tpre = p->Tpre;
    const int fl   = p->fl_pre[b];

    float tacc = 0.0f;
    for (int s = tid; s < S; s += 256) {
        float cc = cB[s];
        float cp = (s > 0) ? cB[s - 1] : 0.0f;
        int right = imin(__float2int_rn(cc), Tpre);
        int left  = imin(__float2int_rn(cp), Tpre);
        int fire  = right - left;
        float alpha = cc - cp;
        float rw = (fire > 0) ? (cc - (float)right * BETA_F) : 0.0f;
        float lw = alpha - rw - (float)imax(fire - 1, 0) * BETA_F;
        if (fire > 0 && right == fl) tacc += rw;
        if (left == fl)              tacc += lw;
    }
    __shared__ float sd[256];
    sd[tid] = tacc;
    __syncthreads();
    for (int off = 128; off > 0; off >>= 1) {
        if (tid < off) sd[tid] += sd[tid + off];
        __syncthreads();
    }
    if (tid == 0) {
        float tw = sd[0];
        int extend = (tw >= TAIL_THRES_F) ? 1 : 0;
        float scale = extend ? (BETA_F / tw) : 1.0f;
        int flf = fl + extend;
        p->scale[b]    = scale;
        p->fl_final[b] = flf;
        lens_out[b] = (float)imax(flf, 1);
    }
}

// ---------------- Kernel 3: banded GEMM via V_WMMA_F32_16X16X4_F32 ---------
// block = (b, frame-tile of 16). 256 threads = 8 waves; wave w owns channel
// columns [w*64, w*64+64) as four 16x16 output tiles. Requires C == 512.
__global__ __launch_bounds__(256) void cif_wmma(const float* __restrict__ src,
                                                const float* __restrict__ csum,
                                                const CifParams* __restrict__ p,
                                                float* __restrict__ out,
                                                int S, int C, int Tout) {
    const int b  = blockIdx.y;
    const int t0 = blockIdx.x * 16;
    const float* cB = csum + (size_t)b * S;
    const float* sB = src + (size_t)b * S * C;

    __shared__ float wtile[16 * WSTRIDE];
    __shared__ float cs_s[KC + 1];
    __shared__ int   sh_lo, sh_hi, sh_Tpre, sh_flpre, sh_flfin;
    __shared__ float sh_total, sh_scale;

    const int tid = threadIdx.x;
    if (tid == 0) {
        int Tpre = p->Tpre;
        float total = p->total[b];
        // first s with csum[s] >= t0 - 0.5  (some right_idx reaches this tile)
        float vlo = (float)t0 - 0.5f;
        int lo = 0, hi = S;
        while (lo < hi) { int mid = (lo + hi) >> 1; if (cB[mid] >= vlo) hi = mid; else lo = mid + 1; }
        int s_lo = lo;
        // first s with csum[s] > t0+15.5 ; +1 => first s with csum[s-1] past tile
        float vhi = (float)t0 + 15.5f;
        lo = 0; hi = S;
        while (lo < hi) { int mid = (lo + hi) >> 1; if (cB[mid] > vhi) hi = mid; else lo = mid + 1; }
        int hiA = lo + 1;
        // last source carrying mass (csum flat afterwards -> zero weights)
        float vt = total - 1e-6f;
        lo = 0; hi = S;
        while (lo < hi) { int mid = (lo + hi) >> 1; if (cB[mid] >= vt) hi = mid; else lo = mid + 1; }
        int s_hi = imin(lo + 1, S);
        if (t0 + 15 < Tpre) s_hi = imin(s_hi, hiA);  // clipped frame T gathers tail
        if (s_lo > s_hi) s_lo = s_hi;
        sh_lo = s_lo; sh_hi = s_hi;
        sh_total = total; sh_Tpre = Tpre;
        sh_flpre = p->fl_pre[b]; sh_flfin = p->fl_final[b]; sh_scale = p->scale[b];
    }
    __syncthreads();
    const int s_lo = sh_lo, s_hi = sh_hi, Tpre = sh_Tpre;
    const float total = sh_total;

    const int lane = tid & 31;
    const int wave = tid >> 5;
    const int half = lane >> 4;         // 0: K=0/1 rows, 1: K=2/3 rows
    const int m    = lane & 15;         // output row / column within tile
    const int kb   = half * 2;
    const float* pwave = sB + wave * 64 + m;   // this wave's channel base

    v8f acc0 = {}, acc1 = {}, acc2 = {}, acc3 = {};

    for (int k0 = s_lo; k0 < s_hi; k0 += KC) {
        __syncthreads();                 // previous chunk fully consumed
        // stage csum window (one extra left element for csum[s-1])
        for (int i = tid; i <= KC; i += 256) {
            int s = k0 - 1 + i;
            cs_s[i] = (s < 0) ? 0.0f : ((s >= S) ? total : cB[s]);
        }
        __syncthreads();
        // build 16 x KC weight band in LDS (m-major, padded stride)
#pragma unroll
        for (int j = 0; j < 4; ++j) {
            int f  = tid * 4 + j;        // 256*4 == 16*KC
            int mm = f >> 6;             // / KC
            int k  = f & (KC - 1);
            int s  = k0 + k;
            float w = 0.0f;
            if (s < s_hi) {
                float cp = cs_s[k], cc = cs_s[k + 1];
                int right = imin(__float2int_rn(cc), Tpre);
                int left  = imin(__float2int_rn(cp), Tpre);
                int fire  = right - left;
                float alpha = cc - cp;
                float rw = (fire > 0) ? (cc - (float)right * BETA_F) : 0.0f;
                int extra = imax(fire - 1, 0);
                float lw = alpha - rw - (float)extra * BETA_F;
                int t = t0 + mm;
                if (t == left)              w += lw;
                if (fire > 0 && t == right) w += rw;
                if (extra > 0) {
                    if (t < Tpre) {
                        int d = t - left;
                        if (d >= 1 && d <= extra) w += BETA_F;
                    } else if (t == Tpre) {  // clipped intermediate fires pile on T
                        int kmin = imax(Tpre - left, 1);
                        int cnt = extra - kmin + 1;
                        if (cnt > 0) w += (float)cnt * BETA_F;
                    }
                }
            }
            wtile[mm * WSTRIDE + k] = w;
        }
        __syncthreads();

        // prefetch next src chunk while we crunch this one
        if ((k0 + KC) < s_hi)
            __builtin_prefetch(&pwave[(size_t)(k0 + KC) * C], 0, 1);

        const int kmax = imin(KC, s_hi - k0);

        // ---- software-pipelined WMMA loop (no divergence: EXEC all-ones) ----
        // fragment loader for k-step kk
        auto ldA = [&](int kk) -> v2f {
            return *(const v2f*)&wtile[m * WSTRIDE + kk + kb];
        };
        int s0 = k0 + kb;
        const float* pA = pwave + (size_t)imin(s0,     S - 1) * C;
        const float* pB = pwave + (size_t)imin(s0 + 1, S - 1) * C;
        v2f a  = ldA(0);
        v2f b0 = { pA[0],  pB[0]  };
        v2f b1 = { pA[16], pB[16] };
        v2f b2 = { pA[32], pB[32] };
        v2f b3 = { pA[48], pB[48] };
        for (int kk = 0; kk < kmax; kk += 4) {
            v2f ca = a, c0v = b0, c1v = b1, c2v = b2, c3v = b3;
            int kn = kk + 4;
            if (kn < kmax) {            // issue next fragments before computing
                a  = ldA(kn);
                int sn = k0 + kn + kb;
                const float* qA = pwave + (size_t)imin(sn,     S - 1) * C;
                const float* qB = pwave + (size_t)imin(sn + 1, S - 1) * C;
                b0 = v2f{ qA[0],  qB[0]  };
                b1 = v2f{ qA[16], qB[16] };
                b2 = v2f{ qA[32], qB[32] };
                b3 = v2f{ qA[48], qB[48] };
            }
            acc0 = __builtin_amdgcn_wmma_f32_16x16x4_f32(false, ca, false, c0v, (short)0, acc0, false, false);
            acc1 = __builtin_amdgcn_wmma_f32_16x16x4_f32(false, ca, false, c1v, (short)0, acc1, false, false);
            acc2 = __builtin_amdgcn_wmma_f32_16x16x4_f32(false, ca, false, c2v, (short)0, acc2, false, false);
            acc3 = __builtin_amdgcn_wmma_f32_16x16x4_f32(false, ca, false, c3v, (short)0, acc3, false, false);
        }
    }

    // store with tail rescale + tail zeroing
    const int flpre = sh_flpre, flfin = sh_flfin;
    const float scale = sh_scale;
    float* pout = out + (size_t)b * Tout * C + wave * 64 + m;
#pragma unroll
    for (int j = 0; j < 8; ++j) {
        int row = t0 + j + half * 8;     // D layout: VGPR j -> M=j / M=8+j
        if (row < Tout) {
            float s0 = (row == flpre) ? scale : 1.0f;
            if (row >= flfin) s0 = 0.0f;
            float* prow = pout + (size_t)row * C;
            prow[0]  = acc0[j] * s0;
            prow[16] = acc1[j] * s0;
            prow[32] = acc2[j] * s0;
            prow[48] = acc3[j] * s0;
        }
    }
}

extern "C" void kernel_launch(void* const* d_in, const int* in_sizes, int n_in,
                              void* d_out, int out_size, void* d_ws, size_t ws_size,
                              hipStream_t stream) {
    const float* src            = (const float*)d_in[0];
    const unsigned char* mask   = (const unsigned char*)d_in[1];
    const float* alphas         = (const float*)d_in[2];
    float* out                  = (float*)d_out;

    const int B  = NB;
    const int BS = in_sizes[1];
    const int S  = BS / B;                     // 4096
    const int C  = in_sizes[0] / BS;           // 512
    int Tout = (out_size - B) / (B * C);       // data-dependent T, fixed by harness
    if (Tout < 1) Tout = 1;

    float* csum  = (float*)d_ws;
    CifParams* p = (CifParams*)((char*)d_ws + (size_t)B * S * sizeof(float));

    cif_init<<<1, 1, 0, stream>>>(p);
    cif_scan<<<B, 256, 0, stream>>>(alphas, mask, csum, p, S);
    cif_tail<<<B, 256, 0, stream>>>(csum, p, out + (size_t)B * Tout * C, S);

    dim3 grid((Tout + 15) / 16, B);
    cif_wmma<<<grid, 256, 0, stream>>>(src, csum, p, out, S, C, Tout);
}